// ChamferLoss_8830452761307
// MI455X (gfx1250) — compile-verified
//
#include <hip/hip_runtime.h>
#include <math.h>

typedef __attribute__((ext_vector_type(2))) float v2f;
typedef __attribute__((ext_vector_type(8))) float v8f;

#define INF_F __builtin_inff()

// ---------------------------------------------------------------------------
// Pack recon points as (x, y, z, |p|^2) for 16B-aligned b128 loads.
// ---------------------------------------------------------------------------
__global__ void pack_points_kernel(const float* __restrict__ pts,
                                   float4* __restrict__ out, int total) {
    int i = blockIdx.x * blockDim.x + threadIdx.x;
    if (i < total) {
        const float x = pts[3 * i], y = pts[3 * i + 1], z = pts[3 * i + 2];
        out[i] = make_float4(x, y, z, fmaf(x, x, fmaf(y, y, z * z)));
    }
}

// ---------------------------------------------------------------------------
// Apply per-batch (R, t) to src points and pack as (gx, gy, gz, |g|^2).
// gts[n,k] = sum_d src[n,d] * R[k,d] + t[k]   (R = transform[:3,:3] rows)
// ---------------------------------------------------------------------------
__global__ void pack_gts_kernel(const float* __restrict__ src,
                                const float* __restrict__ transform,
                                float4* __restrict__ out, int N, int total) {
    int i = blockIdx.x * blockDim.x + threadIdx.x;
    if (i < total) {
        const int b = i / N;
        const float* T = transform + (size_t)b * 16;
        const float sx = src[3 * i], sy = src[3 * i + 1], sz = src[3 * i + 2];
        const float gx = fmaf(T[0], sx, fmaf(T[1], sy, fmaf(T[2],  sz, T[3])));
        const float gy = fmaf(T[4], sx, fmaf(T[5], sy, fmaf(T[6],  sz, T[7])));
        const float gz = fmaf(T[8], sx, fmaf(T[9], sy, fmaf(T[10], sz, T[11])));
        out[i] = make_float4(gx, gy, gz, fmaf(gx, gx, fmaf(gy, gy, gz * gz)));
    }
}

// ---------------------------------------------------------------------------
// Generic directed-Chamfer kernel: one wave32 per (batch, 32-row A tile).
// Sweeps all 16-col B tiles with V_WMMA_F32_16X16X4_F32 (2 WMMAs / B tile):
//   A row i: ( ax, ay, az, |a|^2 )     (from packed float4, built lane-local)
//   B col j: (-2bx,-2by,-2bz, 1 )
//   D[i,j] = -2 a.b + |a|^2 ;  P[i,j] = D[i,j] + |b|^2  (lane-local q.w)
// Running row-mins stay in registers; xor-shuffle reduced; per-block partial
// sum written race-free.  No atomics, no in-loop divergence -> EXEC stays
// all-ones around every WMMA.
// ---------------------------------------------------------------------------
__global__ void __launch_bounds__(32)
chamfer_min_kernel(const float4* __restrict__ Apts,   // (B, nA) packed
                   const float4* __restrict__ Bpts,   // (B, nB) packed
                   float* __restrict__ partials,      // one slot per block
                   int nA, int nB) {
    const int tilesPerBatch = nA >> 5;                // 32 A rows per block
    const int blk  = blockIdx.x;
    const int b    = blk / tilesPerBatch;
    const int ti   = blk - b * tilesPerBatch;
    const int lane = threadIdx.x;
    const int row  = lane & 15;
    const bool hi  = lane >= 16;

    // Two A matrices (16x4 f32 layout): lanes 0-15 hold (K0,K1)=(x,y) of
    // row=lane; lanes 16-31 hold (K2,K3)=(z,|a|^2) of row=lane-16.
    const float4* Abase = Apts + (size_t)b * nA + (size_t)ti * 32;
    const float4 p0 = Abase[row];
    const float4 p1 = Abase[16 + row];
    v2f a0, a1;
    a0.x = hi ? p0.z : p0.x;  a0.y = hi ? p0.w : p0.y;
    a1.x = hi ? p1.z : p1.x;  a1.y = hi ? p1.w : p1.y;

    float rmin0[8], rmin1[8];
#pragma unroll
    for (int v = 0; v < 8; ++v) { rmin0[v] = INF_F; rmin1[v] = INF_F; }

    const float4* Bbase = Bpts + (size_t)b * nB;
    const int bTiles = nB >> 4;
#pragma unroll 2
    for (int bt = 0; bt < bTiles; ++bt) {
        const float4 q = Bbase[bt * 16 + row];        // global_load_b128
        // B-matrix 4x16: VGPR0 = rows K0 (lanes 0-15) / K2 (lanes 16-31),
        //                VGPR1 = rows K1 / K3; column index = lane%16.
        v2f bm;
        bm.x = hi ? (-2.0f * q.z) : (-2.0f * q.x);
        bm.y = hi ? 1.0f          : (-2.0f * q.y);
        const float yy = q.w;                         // lane's own column

        v8f c = {};
        v8f d0 = __builtin_amdgcn_wmma_f32_16x16x4_f32(
            false, a0, false, bm, (short)0, c, false, false);
        v8f d1 = __builtin_amdgcn_wmma_f32_16x16x4_f32(
            false, a1, false, bm, (short)0, c, false, false);

#pragma unroll
        for (int v = 0; v < 8; ++v) {
            rmin0[v] = fminf(rmin0[v], d0[v] + yy);
            rmin1[v] = fminf(rmin1[v], d1[v] + yy);
        }
    }

    // Row-min reduction: within each 16-lane half, lanes hold different
    // columns of the same row; xor {1,2,4,8} stays inside the half.
    float rs = 0.0f;
#pragma unroll
    for (int v = 0; v < 8; ++v) {
        float r0 = rmin0[v], r1 = rmin1[v];
        r0 = fminf(r0, __shfl_xor(r0, 1, 32));
        r0 = fminf(r0, __shfl_xor(r0, 2, 32));
        r0 = fminf(r0, __shfl_xor(r0, 4, 32));
        r0 = fminf(r0, __shfl_xor(r0, 8, 32));
        r1 = fminf(r1, __shfl_xor(r1, 1, 32));
        r1 = fminf(r1, __shfl_xor(r1, 2, 32));
        r1 = fminf(r1, __shfl_xor(r1, 4, 32));
        r1 = fminf(r1, __shfl_xor(r1, 8, 32));
        rs += r0 + r1;        // lo lanes: rows 0-7; hi lanes: rows 8-15
    }
    rs += __shfl_xor(rs, 16, 32);                     // all 32 rows of block
    if (lane == 0) partials[blk] = rs;
}

// ---------------------------------------------------------------------------
// Deterministic final sum of all per-block partials (both passes).
// ---------------------------------------------------------------------------
__global__ void __launch_bounds__(256)
chamfer_finalize(const float* __restrict__ partials, int count,
                 float* __restrict__ out) {
    __shared__ float sdata[256];
    float s = 0.0f;
    for (int i = threadIdx.x; i < count; i += 256) s += partials[i];
    sdata[threadIdx.x] = s;
    __syncthreads();
    for (int off = 128; off > 0; off >>= 1) {
        if (threadIdx.x < off) sdata[threadIdx.x] += sdata[threadIdx.x + off];
        __syncthreads();
    }
    if (threadIdx.x == 0) out[0] = sdata[0];
}

extern "C" void kernel_launch(void* const* d_in, const int* in_sizes, int n_in,
                              void* d_out, int out_size, void* d_ws, size_t ws_size,
                              hipStream_t stream) {
    const float* recon     = (const float*)d_in[0]; // (B,M,3) preds
    const float* src       = (const float*)d_in[1]; // (B,N,3)
    const float* transform = (const float*)d_in[2]; // (B,4,4)

    const int B = in_sizes[2] / 16;
    const int M = in_sizes[0] / (3 * B);
    const int N = in_sizes[1] / (3 * B);

    // Workspace layout (16B-aligned chunks)
    float4* packR = (float4*)d_ws;                  // B*M packed preds
    float4* packG = packR + (size_t)B * M;          // B*N packed gts
    float*  parts = (float*)(packG + (size_t)B * N);

    const int totR = B * M;
    const int totG = B * N;
    const int blocks2 = B * (N / 32);               // loss_2: per-gts min over preds
    const int blocks1 = B * (M / 32);               // loss_1: per-pred min over gts

    pack_points_kernel<<<(totR + 255) / 256, 256, 0, stream>>>(recon, packR, totR);
    pack_gts_kernel<<<(totG + 255) / 256, 256, 0, stream>>>(src, transform, packG,
                                                            N, totG);

    // loss_2: rows = gts, sweep preds
    chamfer_min_kernel<<<blocks2, 32, 0, stream>>>(packG, packR, parts, N, M);
    // loss_1: rows = preds, sweep gts
    chamfer_min_kernel<<<blocks1, 32, 0, stream>>>(packR, packG, parts + blocks2,
                                                   M, N);

    chamfer_finalize<<<1, 256, 0, stream>>>(parts, blocks2 + blocks1,
                                            (float*)d_out);
}